// ContagionGAT_65807488909417
// MI455X (gfx1250) — compile-verified
//
#include <hip/hip_runtime.h>
#include <hip/hip_bf16.h>
#include <math.h>

#define HEADS 4
#define HID   32
#define HC    128
#define NEG_SLOPE 0.2f

typedef float v2f __attribute__((ext_vector_type(2)));
typedef float v8f __attribute__((ext_vector_type(8)));

// ---------------------------------------------------------------------------
// float atomic max via signed/unsigned int atomics (works for mixed signs).
// amax buffers are pre-seeded with 0xFFFFFFFF bytes: as signed int it's -1
// (any non-negative float beats it), as unsigned it's UINT_MAX (any negative
// float beats it). Every node has a self-loop, so amax always gets written.
// ---------------------------------------------------------------------------
__device__ __forceinline__ void atomicMaxFloat(float* addr, float val) {
    if (val >= 0.0f) {
        atomicMax((int*)addr, __float_as_int(val));
    } else {
        atomicMin((unsigned int*)addr, __float_as_uint(val));
    }
}

// ---------------------------------------------------------------------------
// WMMA fp32 GEMM: C[M x Ncols] = A[M x K] @ B[K x Ncols].
// blockDim = 256 (8 waves). blockIdx.x -> 16-row tile. wave -> 16-col tile.
// A-frag (16x4 f32): lane l: m=l&15, hi=l>>4; VGPR0 = A[m][k+2*hi],
// VGPR1 = A[m][k+2*hi+1].  B-frag (4x16) mirrored.  D per ISA layout.
// ---------------------------------------------------------------------------
__global__ void gemm_wmma_f32(const float* __restrict__ A,
                              const float* __restrict__ B,
                              float* __restrict__ C,
                              int M, int K, int Ncols) {
    const int wave = threadIdx.x >> 5;
    const int lane = threadIdx.x & 31;
    const int m  = lane & 15;
    const int hi = lane >> 4;
    const int row0 = blockIdx.x * 16;
    const int col0 = wave * 16;
    if (col0 >= Ncols) return;               // wave-uniform exit

    int rA = row0 + m;
    if (rA >= M) rA = M - 1;                 // clamp loads, guard stores below

    v8f acc = {};
    for (int k = 0; k < K; k += 4) {
        v2f a, b;
        const float* ap = A + (size_t)rA * K + (size_t)(k + 2 * hi);
        a.x = ap[0];
        a.y = ap[1];
        const float* bp = B + (size_t)(k + 2 * hi) * Ncols + col0 + m;
        b.x = bp[0];
        b.y = bp[Ncols];
        acc = __builtin_amdgcn_wmma_f32_16x16x4_f32(
            false, a, false, b, (short)0, acc, false, false);
    }
#pragma unroll
    for (int v = 0; v < 8; ++v) {
        int r = row0 + v + 8 * hi;
        if (r < M) C[(size_t)r * Ncols + col0 + m] = acc[v];
    }
}

// ---------------------------------------------------------------------------
// misc[] layout: [0] = mean(edge_attr); [1..4] = ecoef layer1; [5..8] = layer2
// ---------------------------------------------------------------------------
__global__ void mean_reduce(const float* __restrict__ ea, float* __restrict__ misc, int E) {
    __shared__ float sm[256];
    int t = threadIdx.x;
    float s = 0.0f;
    for (int i = blockIdx.x * 256 + t; i < E; i += gridDim.x * 256) s += ea[i];
    sm[t] = s;
    __syncthreads();
    for (int o = 128; o > 0; o >>= 1) {
        if (t < o) sm[t] += sm[t + o];
        __syncthreads();
    }
    if (t == 0) atomicAdd(&misc[0], sm[0] / (float)E);
}

__global__ void prep_coef(const float* __restrict__ We1, const float* __restrict__ ae1,
                          const float* __restrict__ We2, const float* __restrict__ ae2,
                          float* __restrict__ misc) {
    int h = threadIdx.x;                     // 0..7
    if (h < HEADS) {
        float s = 0.0f;
        for (int c = 0; c < HID; ++c) s += We1[h * HID + c] * ae1[h * HID + c];
        misc[1 + h] = s;
    } else if (h < 2 * HEADS) {
        int hh = h - HEADS;
        float s = 0.0f;
        for (int c = 0; c < HID; ++c) s += We2[hh * HID + c] * ae2[hh * HID + c];
        misc[5 + hh] = s;
    }
}

// al_s / al_d : per-(node, head) 32-length dots against a_src / a_dst.
__global__ void node_alpha(const float* __restrict__ h,
                           const float* __restrict__ a_s,
                           const float* __restrict__ a_d,
                           float* __restrict__ als, float* __restrict__ ald, int Nn) {
    int t = blockIdx.x * blockDim.x + threadIdx.x;
    if (t >= Nn * HEADS) return;
    int n = t / HEADS, hd = t % HEADS;
    const float* hp = h + (size_t)n * HC + hd * HID;
    const float* sp = a_s + hd * HID;
    const float* dp = a_d + hd * HID;
    float s = 0.0f, d = 0.0f;
#pragma unroll
    for (int c = 0; c < HID; ++c) {
        float v = hp[c];
        s += v * sp[c];
        d += v * dp[c];
    }
    als[t] = s;
    ald[t] = d;
}

// Edge pass A (H=4): alpha = leaky_relu(al_s[src]+al_d[dst]+ec[h]*ea); seg-max.
__global__ void edge_alpha4(const int* __restrict__ ei, const float* __restrict__ eattr,
                            const float* __restrict__ misc, int eoff,
                            const float* __restrict__ als, const float* __restrict__ ald,
                            float* __restrict__ alpha, float* __restrict__ amax,
                            int E, int Nn) {
    int e = blockIdx.x * blockDim.x + threadIdx.x;
    int E2 = E + Nn;
    if (e >= E2) return;
    int s, d;
    float ea;
    if (e < E) { s = ei[e]; d = ei[E + e]; ea = eattr[e]; }
    else       { s = d = e - E;            ea = misc[0]; }
#pragma unroll
    for (int h = 0; h < HEADS; ++h) {
        float a = als[s * HEADS + h] + ald[d * HEADS + h] + misc[eoff + h] * ea;
        a = (a > 0.0f) ? a : NEG_SLOPE * a;
        alpha[(size_t)e * HEADS + h] = a;
        atomicMaxFloat(&amax[d * HEADS + h], a);
    }
}

// Edge pass B (H=4): ex = exp(alpha - amax[dst]); seg-sum into den.
__global__ void edge_exp4(const int* __restrict__ ei, float* __restrict__ alpha,
                          const float* __restrict__ amax, float* __restrict__ den,
                          int E, int Nn) {
    int e = blockIdx.x * blockDim.x + threadIdx.x;
    int E2 = E + Nn;
    if (e >= E2) return;
    int d = (e < E) ? ei[E + e] : (e - E);
#pragma unroll
    for (int h = 0; h < HEADS; ++h) {
        float ex = expf(alpha[(size_t)e * HEADS + h] - amax[d * HEADS + h]);
        alpha[(size_t)e * HEADS + h] = ex;
        atomicAdd(&den[d * HEADS + h], ex);
    }
}

// Edge pass C (H=4): one wave32 per edge; lane j*32+lane covers channel space,
// head index == j.  agg[dst, c] += h[src, c] * att[head].
__global__ void edge_aggr4(const int* __restrict__ ei, const float* __restrict__ ex,
                           const float* __restrict__ den, const float* __restrict__ hf,
                           float* __restrict__ agg, int E, int Nn) {
    int gt = blockIdx.x * blockDim.x + threadIdx.x;
    int wid = gt >> 5;
    int lane = gt & 31;
    int E2 = E + Nn;
    if (wid >= E2) return;
    int s, d;
    if (wid < E) { s = ei[wid]; d = ei[E + wid]; }
    else         { s = d = wid - E; }
#pragma unroll
    for (int j = 0; j < HEADS; ++j) {
        int c = j * 32 + lane;               // head = j
        float att = ex[(size_t)wid * HEADS + j] / (den[d * HEADS + j] + 1e-16f);
        atomicAdd(&agg[(size_t)d * HC + c], hf[(size_t)s * HC + c] * att);
    }
}

// out = ELU(agg + b), elementwise over N*HC.
__global__ void bias_elu(const float* __restrict__ agg, const float* __restrict__ b,
                         float* __restrict__ out, int total) {
    int i = blockIdx.x * blockDim.x + threadIdx.x;
    if (i >= total) return;
    float v = agg[i] + b[i & (HC - 1)];
    out[i] = (v > 0.0f) ? v : (expf(v) - 1.0f);
}

// ------------------------- Layer 3 (H=1, C=1) ------------------------------
__global__ void matvec128(const float* __restrict__ h, const float* __restrict__ W3,
                          float* __restrict__ h3, int Nn) {
    int n = blockIdx.x * blockDim.x + threadIdx.x;
    if (n >= Nn) return;
    float s = 0.0f;
#pragma unroll 8
    for (int c = 0; c < HC; ++c) s += h[(size_t)n * HC + c] * W3[c];
    h3[n] = s;
}

__global__ void edge_alpha1(const int* __restrict__ ei, const float* __restrict__ eattr,
                            const float* __restrict__ misc, const float* __restrict__ h3,
                            const float* __restrict__ as3, const float* __restrict__ ad3,
                            const float* __restrict__ we3, const float* __restrict__ ae3,
                            float* __restrict__ alpha, float* __restrict__ amax,
                            int E, int Nn) {
    int e = blockIdx.x * blockDim.x + threadIdx.x;
    int E2 = E + Nn;
    if (e >= E2) return;
    int s, d;
    float ea;
    if (e < E) { s = ei[e]; d = ei[E + e]; ea = eattr[e]; }
    else       { s = d = e - E;            ea = misc[0]; }
    float a = as3[0] * h3[s] + ad3[0] * h3[d] + we3[0] * ae3[0] * ea;
    a = (a > 0.0f) ? a : NEG_SLOPE * a;
    alpha[e] = a;
    atomicMaxFloat(&amax[d], a);
}

__global__ void edge_exp1(const int* __restrict__ ei, float* __restrict__ alpha,
                          const float* __restrict__ amax, float* __restrict__ den,
                          int E, int Nn) {
    int e = blockIdx.x * blockDim.x + threadIdx.x;
    int E2 = E + Nn;
    if (e >= E2) return;
    int d = (e < E) ? ei[E + e] : (e - E);
    float ex = expf(alpha[e] - amax[d]);
    alpha[e] = ex;
    atomicAdd(&den[d], ex);
}

__global__ void edge_aggr1(const int* __restrict__ ei, const float* __restrict__ ex,
                           const float* __restrict__ den, const float* __restrict__ h3,
                           float* __restrict__ out, int E, int Nn) {
    int e = blockIdx.x * blockDim.x + threadIdx.x;
    int E2 = E + Nn;
    if (e >= E2) return;
    int s, d;
    if (e < E) { s = ei[e]; d = ei[E + e]; }
    else       { s = d = e - E; }
    float att = ex[e] / (den[d] + 1e-16f);
    atomicAdd(&out[d], h3[s] * att);
}

__global__ void add_bias1(float* __restrict__ out, const float* __restrict__ b3, int Nn) {
    int n = blockIdx.x * blockDim.x + threadIdx.x;
    if (n >= Nn) return;
    out[n] = out[n] + b3[0];
}

// ---------------------------------------------------------------------------
extern "C" void kernel_launch(void* const* d_in, const int* in_sizes, int n_in,
                              void* d_out, int out_size, void* d_ws, size_t ws_size,
                              hipStream_t stream) {
    const float* x      = (const float*)d_in[0];
    const int*   ei     = (const int*)  d_in[1];
    const float* eattr  = (const float*)d_in[2];
    const float* W1     = (const float*)d_in[3];
    const float* a_src1 = (const float*)d_in[4];
    const float* a_dst1 = (const float*)d_in[5];
    const float* We1    = (const float*)d_in[6];
    const float* a_e1   = (const float*)d_in[7];
    const float* b1     = (const float*)d_in[8];
    const float* W2     = (const float*)d_in[9];
    const float* a_src2 = (const float*)d_in[10];
    const float* a_dst2 = (const float*)d_in[11];
    const float* We2    = (const float*)d_in[12];
    const float* a_e2   = (const float*)d_in[13];
    const float* b2     = (const float*)d_in[14];
    const float* W3     = (const float*)d_in[15];
    const float* a_src3 = (const float*)d_in[16];
    const float* a_dst3 = (const float*)d_in[17];
    const float* We3    = (const float*)d_in[18];
    const float* a_e3   = (const float*)d_in[19];
    const float* b3     = (const float*)d_in[20];
    (void)n_in; (void)out_size; (void)ws_size;

    const int N  = in_sizes[0] / 4;          // F_IN = 4
    const int E  = in_sizes[1] / 2;
    const int E2 = E + N;

    // ---- workspace layout (256B aligned) ----
    char* ws = (char*)d_ws;
    size_t off = 0;
    auto alloc = [&](size_t bytes) -> float* {
        float* p = (float*)(ws + off);
        off = (off + bytes + 255) & ~(size_t)255;
        return p;
    };
    float* misc  = alloc(256);                          // mean + ecoefs
    float* HA    = alloc((size_t)N * HC * 4);           // linear outputs
    float* HB    = alloc((size_t)N * HC * 4);           // activated features
    float* AGG   = alloc((size_t)N * HC * 4);           // aggregation accum
    float* ALPHA = alloc((size_t)E2 * HEADS * 4);       // alpha / ex
    float* ALS   = alloc((size_t)N * HEADS * 4);
    float* ALD   = alloc((size_t)N * HEADS * 4);
    float* AMAX  = alloc((size_t)N * HEADS * 4);
    float* DEN   = alloc((size_t)N * HEADS * 4);
    float* H3    = alloc((size_t)N * 4);

    const int TB = 256;
    const int gEdge  = (E2 + TB - 1) / TB;
    const int gEdgeW = (E2 * 32 + TB - 1) / TB;         // wave-per-edge
    const int gNH    = (N * HEADS + TB - 1) / TB;
    const int gNC    = (N * HC + TB - 1) / TB;
    const int gN     = (N + TB - 1) / TB;
    const int gTile  = (N + 15) / 16;
    int gMean = (E + TB - 1) / TB; if (gMean > 1024) gMean = 1024;

    // ---- precompute: mean(edge_attr), edge coefficients ----
    hipMemsetAsync(misc, 0, 256, stream);
    mean_reduce<<<gMean, TB, 0, stream>>>(eattr, misc, E);
    prep_coef<<<1, 8, 0, stream>>>(We1, a_e1, We2, a_e2, misc);

    // ================= Layer 1: GAT(4 -> 32, H=4) =================
    gemm_wmma_f32<<<gTile, TB, 0, stream>>>(x, W1, HA, N, 4, HC);
    node_alpha<<<gNH, TB, 0, stream>>>(HA, a_src1, a_dst1, ALS, ALD, N);
    hipMemsetAsync(AMAX, 0xFF, (size_t)N * HEADS * 4, stream);
    hipMemsetAsync(DEN,  0,    (size_t)N * HEADS * 4, stream);
    hipMemsetAsync(AGG,  0,    (size_t)N * HC * 4,    stream);
    edge_alpha4<<<gEdge, TB, 0, stream>>>(ei, eattr, misc, 1, ALS, ALD, ALPHA, AMAX, E, N);
    edge_exp4<<<gEdge, TB, 0, stream>>>(ei, ALPHA, AMAX, DEN, E, N);
    edge_aggr4<<<gEdgeW, TB, 0, stream>>>(ei, ALPHA, DEN, HA, AGG, E, N);
    bias_elu<<<gNC, TB, 0, stream>>>(AGG, b1, HB, N * HC);

    // ================= Layer 2: GAT(128 -> 32, H=4) =================
    gemm_wmma_f32<<<gTile, TB, 0, stream>>>(HB, W2, HA, N, HC, HC);
    node_alpha<<<gNH, TB, 0, stream>>>(HA, a_src2, a_dst2, ALS, ALD, N);
    hipMemsetAsync(AMAX, 0xFF, (size_t)N * HEADS * 4, stream);
    hipMemsetAsync(DEN,  0,    (size_t)N * HEADS * 4, stream);
    hipMemsetAsync(AGG,  0,    (size_t)N * HC * 4,    stream);
    edge_alpha4<<<gEdge, TB, 0, stream>>>(ei, eattr, misc, 5, ALS, ALD, ALPHA, AMAX, E, N);
    edge_exp4<<<gEdge, TB, 0, stream>>>(ei, ALPHA, AMAX, DEN, E, N);
    edge_aggr4<<<gEdgeW, TB, 0, stream>>>(ei, ALPHA, DEN, HA, AGG, E, N);
    bias_elu<<<gNC, TB, 0, stream>>>(AGG, b2, HB, N * HC);

    // ================= Layer 3: GAT(128 -> 1, H=1) =================
    matvec128<<<gN, TB, 0, stream>>>(HB, W3, H3, N);
    hipMemsetAsync(AMAX, 0xFF, (size_t)N * 4, stream);
    hipMemsetAsync(DEN,  0,    (size_t)N * 4, stream);
    hipMemsetAsync(d_out, 0,   (size_t)N * 4, stream);
    edge_alpha1<<<gEdge, TB, 0, stream>>>(ei, eattr, misc, H3, a_src3, a_dst3, We3, a_e3,
                                          ALPHA, AMAX, E, N);
    edge_exp1<<<gEdge, TB, 0, stream>>>(ei, ALPHA, AMAX, DEN, E, N);
    edge_aggr1<<<gEdge, TB, 0, stream>>>(ei, ALPHA, DEN, H3, (float*)d_out, E, N);
    add_bias1<<<gN, TB, 0, stream>>>((float*)d_out, b3, N);
}